// TripletLoss_28054726377830
// MI455X (gfx1250) — compile-verified
//
#include <hip/hip_runtime.h>
#include <stdint.h>

#define B_N    16384
#define D_K    128
#define TILE   128
#define KT     64     // K half staged per LDS buffer
#define KSTR   68     // padded LDS row stride (floats): 68 % 64 == 4 -> conflict-free b64 frag reads
#define EPS    1e-6f
#define MARGIN 0.3f

typedef __attribute__((ext_vector_type(2))) float v2f;
typedef __attribute__((ext_vector_type(8))) float v8f;

__device__ __forceinline__ float wave_sum(float v) {
#pragma unroll
    for (int m = 16; m >= 1; m >>= 1) v += __shfl_xor(v, m, 32);
    return v;
}

// Kernel 1: per-row stats. One wave (32 lanes) per row; each lane owns 4 of 128 cols.
__global__ __launch_bounds__(256) void prep_kernel(
    const float* __restrict__ a, const float* __restrict__ p,
    const float* __restrict__ n, float* __restrict__ colterm,
    float* __restrict__ posdist, unsigned long long* __restrict__ rowkey) {
    const int wave = threadIdx.x >> 5, lane = threadIdx.x & 31;
    const int row  = blockIdx.x * 8 + wave;
    const float4 av = *(const float4*)(a + (size_t)row * D_K + lane * 4);
    const float4 pv = *(const float4*)(p + (size_t)row * D_K + lane * 4);
    const float4 nv = *(const float4*)(n + (size_t)row * D_K + lane * 4);
    float d0 = av.x - pv.x + EPS, d1 = av.y - pv.y + EPS;
    float d2 = av.z - pv.z + EPS, d3 = av.w - pv.w + EPS;
    float s  = wave_sum(d0 * d0 + d1 * d1 + d2 * d2 + d3 * d3);
    float n2 = wave_sum(nv.x * nv.x + nv.y * nv.y + nv.z * nv.z + nv.w * nv.w);
    float sn = wave_sum(nv.x + nv.y + nv.z + nv.w);
    if (lane == 0) {
        posdist[row] = sqrtf(s);
        colterm[row] = n2 - 2.0f * EPS * sn;  // row-constant terms of d^2 dropped
        rowkey[row]  = 0xFFFFFFFFFFFFFFFFull; // re-init every call (ws is poisoned once)
    }
}

// Kernel 2: 128x128 tile of cross = A @ N^T via v_wmma_f32_16x16x4_f32, fused row-argmin.
// Tiles staged memory->LDS with GLOBAL_LOAD_ASYNC_TO_LDS_B128 (ASYNCcnt), double-buffered
// over the two K-halves so half-1 streams in while half-0 is consumed by WMMA.
__global__ __launch_bounds__(256) void gemm_argmin_kernel(
    const float* __restrict__ A, const float* __restrict__ Ng,
    const float* __restrict__ colterm, unsigned long long* __restrict__ rowkey) {
    __shared__ __align__(16) float As[2][TILE * KSTR];
    __shared__ __align__(16) float Bs[2][TILE * KSTR];
    const int t    = threadIdx.x;
    const int wave = t >> 5;
    const int lane = t & 31;
    const int half = lane >> 4;     // 0: C rows v ; 1: C rows 8+v  (WMMA C layout)
    const int lm   = lane & 15;     // N index within 16x16 tile
    const int rbase = blockIdx.y * TILE;
    const int cbase = blockIdx.x * TILE;

    // Issue ALL async memory->LDS copies up front (32 instrs/wave; first 16 = buffer 0).
#pragma unroll
    for (int h = 0; h < 2; h++) {
        const int kt = h * KT;
#pragma unroll
        for (int i = 0; i < 8; i++) {
            int idx = i * 256 + t;          // 0..2047
            int r   = idx >> 4;             // 0..127
            int k4  = (idx & 15) << 2;      // 0..60
            unsigned la = (unsigned)(uintptr_t)&As[h][r * KSTR + k4];
            unsigned lb = (unsigned)(uintptr_t)&Bs[h][r * KSTR + k4];
            const float* ga = A  + (size_t)(rbase + r) * D_K + kt + k4;
            const float* gb = Ng + (size_t)(cbase + r) * D_K + kt + k4;
            asm volatile("global_load_async_to_lds_b128 %0, %1, off"
                         :: "v"(la), "v"(ga) : "memory");
            asm volatile("global_load_async_to_lds_b128 %0, %1, off"
                         :: "v"(lb), "v"(gb) : "memory");
        }
    }

    v8f acc[8] = {};  // 8 col-tiles x 8 VGPRs = 64 accumulator VGPRs / lane

#pragma unroll
    for (int h = 0; h < 2; h++) {
        // Async loads complete in order: <=16 outstanding => buffer 0 landed.
        if (h == 0) asm volatile("s_wait_asynccnt 0x10" ::: "memory");
        else        asm volatile("s_wait_asynccnt 0x0"  ::: "memory");
        __syncthreads();  // cross-wave visibility of LDS tiles

        const float* arow = &As[h][(wave * 16 + lm) * KSTR + 2 * half];
        const float* bbas = &Bs[h][lm * KSTR + 2 * half];
#pragma unroll 4
        for (int k = 0; k < KT; k += 4) {
            // A 16x4 frag: lane(0-15)->K=k,k+1 ; lane(16-31)->K=k+2,k+3 (M = lm)
            v2f afrag = *(const v2f*)(arow + k);
#pragma unroll
            for (int ct = 0; ct < 8; ct++) {
                // B 4x16 frag (N^T): same K split, N = lm within col-tile ct
                v2f bfrag = *(const v2f*)(bbas + ct * 16 * KSTR + k);
                acc[ct] = __builtin_amdgcn_wmma_f32_16x16x4_f32(
                    false, afrag, false, bfrag, (short)0, acc[ct], false, false);
            }
        }
        if (h == 0) __syncthreads();  // everyone done reading buf0 before... (buf0 not reused; kept for safety/order)
    }

    // Fused argmin epilogue: key = colterm[col] - 2*cross  (monotone in d^2 per row)
    float cterm[8];
#pragma unroll
    for (int ct = 0; ct < 8; ct++) cterm[ct] = colterm[cbase + ct * 16 + lm];

    const int row_lo = rbase + wave * 16 + half * 8;
#pragma unroll
    for (int v = 0; v < 8; v++) {
        unsigned long long best = 0xFFFFFFFFFFFFFFFFull;
#pragma unroll
        for (int ct = 0; ct < 8; ct++) {
            int col    = cbase + ct * 16 + lm;
            float keyf = __builtin_fmaf(-2.0f, acc[ct][v], cterm[ct]);
            unsigned u = __float_as_uint(keyf);
            u = (u & 0x80000000u) ? ~u : (u | 0x80000000u);  // order-preserving map
            unsigned long long pk = ((unsigned long long)u << 32) | (unsigned)col;
            best = pk < best ? pk : best;   // ties -> smaller col (first occurrence)
        }
#pragma unroll
        for (int m = 1; m <= 8; m <<= 1) {  // reduce across the 16 lanes of this half
            unsigned long long o = __shfl_xor(best, m, 32);
            best = o < best ? o : best;
        }
        if (lm == 0) atomicMin(&rowkey[row_lo + v], best);
    }
}

// Kernel 3: neg_dist recomputed exactly as reference pairwise_dist; per-row loss.
__global__ __launch_bounds__(256) void finalize_kernel(
    const float* __restrict__ a, const float* __restrict__ n,
    const unsigned long long* __restrict__ rowkey,
    const float* __restrict__ posdist, float* __restrict__ loss) {
    const int wave = threadIdx.x >> 5, lane = threadIdx.x & 31;
    const int row  = blockIdx.x * 8 + wave;
    const int col  = (int)(rowkey[row] & 0xFFFFFFFFull);
    const float4 av = *(const float4*)(a + (size_t)row * D_K + lane * 4);
    const float4 nv = *(const float4*)(n + (size_t)col * D_K + lane * 4);
    float d0 = av.x - nv.x + EPS, d1 = av.y - nv.y + EPS;
    float d2 = av.z - nv.z + EPS, d3 = av.w - nv.w + EPS;
    float s  = wave_sum(d0 * d0 + d1 * d1 + d2 * d2 + d3 * d3);
    if (lane == 0) loss[row] = fmaxf(posdist[row] - sqrtf(s) + MARGIN, 0.0f);
}

// Kernel 4: deterministic fixed-order mean reduction (single block).
__global__ __launch_bounds__(256) void reduce_kernel(const float* __restrict__ loss,
                                                     float* __restrict__ out) {
    __shared__ float sm[256];
    float s = 0.0f;
    for (int i = threadIdx.x; i < B_N; i += 256) s += loss[i];
    sm[threadIdx.x] = s;
    __syncthreads();
    for (int off = 128; off > 0; off >>= 1) {
        if (threadIdx.x < (unsigned)off) sm[threadIdx.x] += sm[threadIdx.x + off];
        __syncthreads();
    }
    if (threadIdx.x == 0) out[0] = sm[0] / (float)B_N;
}

extern "C" void kernel_launch(void* const* d_in, const int* in_sizes, int n_in,
                              void* d_out, int out_size, void* d_ws, size_t ws_size,
                              hipStream_t stream) {
    (void)in_sizes; (void)n_in; (void)out_size; (void)ws_size;
    const float* anchor   = (const float*)d_in[0];
    const float* positive = (const float*)d_in[1];
    const float* negative = (const float*)d_in[2];

    char* ws = (char*)d_ws;
    unsigned long long* rowkey = (unsigned long long*)ws;              // 128 KB
    float* colterm = (float*)(ws + 131072);                            //  64 KB
    float* posdist = (float*)(ws + 131072 + 65536);                    //  64 KB
    float* loss    = (float*)(ws + 131072 + 2 * 65536);                //  64 KB

    prep_kernel<<<B_N / 8, 256, 0, stream>>>(anchor, positive, negative,
                                             colterm, posdist, rowkey);
    dim3 grid(B_N / TILE, B_N / TILE);
    gemm_argmin_kernel<<<grid, 256, 0, stream>>>(anchor, negative, colterm, rowkey);
    finalize_kernel<<<B_N / 8, 256, 0, stream>>>(anchor, negative, rowkey, posdist, loss);
    reduce_kernel<<<1, 256, 0, stream>>>(loss, (float*)d_out);
}